// so2_linear_escnn_edge_88656714925178
// MI455X (gfx1250) — compile-verified
//
#include <hip/hip_runtime.h>
#include <hip/hip_bf16.h>

typedef __attribute__((ext_vector_type(2))) float v2f;
typedef __attribute__((ext_vector_type(4))) float v4f;
typedef __attribute__((ext_vector_type(8))) float v8f;

#define WAVES_PER_BLOCK 8
#define EDGES_PER_WAVE 16
#define EDGES_PER_BLOCK 128

// Pair-interleaved weight layouts: W1P[kp][2*col + j] = W1[2*kp+j][col]
// Row strides == 32 (mod 64) floats so the hi-half-wave (K+2) b64 reads land on
// the opposite 32 banks -> conflict-free ds_load_b64 B fragments.
#define W1P_STRIDE 288   // floats; 32 pair-rows, 256 used + 32 pad
#define W2P_STRIDE 96    // floats; 64 pair-rows, 64 used + 32 pad
#define H_STRIDE   132   // per-wave 16 x 128 hidden tile (conflict-free float2 reloads)
#define WW_STRIDE  36    // per-wave 16 x 32 w tile (overlays h region)

#define LDS_W1_OFF 0
#define LDS_W2_OFF (32 * W1P_STRIDE)                      // 9216
#define LDS_B1_OFF (LDS_W2_OFF + 64 * W2P_STRIDE)         // 15360
#define LDS_B2_OFF (LDS_B1_OFF + 128)                     // 15488
#define LDS_H_OFF  (LDS_B2_OFF + 32)                      // 15520
#define LDS_H_PER_WAVE (EDGES_PER_WAVE * H_STRIDE)        // 2112
#define LDS_FLOATS (LDS_H_OFF + WAVES_PER_BLOCK * LDS_H_PER_WAVE)  // 32416 floats = 129664 B

__device__ __forceinline__ v2f lds_v2f(const float* p) {
    return *(const v2f*)__builtin_assume_aligned(p, 8);   // -> ds_load_b64
}

__global__ __launch_bounds__(256) void so2_linear_escnn_kernel(
    const float* __restrict__ x,  const float* __restrict__ s,
    const float* __restrict__ W1, const float* __restrict__ b1,
    const float* __restrict__ W2, const float* __restrict__ b2,
    float* __restrict__ out, int E)
{
    extern __shared__ float lds[];
    const int tid = threadIdx.x;

    // ---- Stage weights/biases into LDS, pair-interleaved over K ----
    for (int i = tid; i < 64 * 128; i += 256) {
        const int k = i >> 7, c = i & 127;
        lds[LDS_W1_OFF + (k >> 1) * W1P_STRIDE + 2 * c + (k & 1)] = W1[i];
    }
    for (int i = tid; i < 128 * 32; i += 256) {
        const int k = i >> 5, c = i & 31;
        lds[LDS_W2_OFF + (k >> 1) * W2P_STRIDE + 2 * c + (k & 1)] = W2[i];
    }
    if (tid < 128) lds[LDS_B1_OFF + tid] = b1[tid];
    if (tid < 32)  lds[LDS_B2_OFF + tid] = b2[tid];
    __syncthreads();

    const int wave  = tid >> 5;
    const int lane  = tid & 31;
    const int laneM = lane & 15;   // M row (A/C) or N column (B/C) within 16-tile
    const int hi    = lane >> 4;   // half-wave selector: K offset +2 in A/B frags

    const int eBase = blockIdx.x * EDGES_PER_BLOCK + wave * EDGES_PER_WAVE;
    int eRow = eBase + laneM;
    if (eRow > E - 1) eRow = E - 1;        // clamp loads; WMMA needs EXEC all-ones

    // ---- GEMM1: h[16x128] = relu(s[16x64] @ W1[64x128] + b1) ----
    // A frag (f32 16x4): lanes 0-15 hold M=lane, K={k0,k0+1}; lanes 16-31 K={k0+2,k0+3}
    v2f a1[16];
    {
        const float* srow = s + (long long)eRow * 64;
        #pragma unroll
        for (int kk = 0; kk < 16; ++kk)
            a1[kk] = *(const v2f*)(srow + 4 * kk + 2 * hi);
    }

    float* hw = lds + LDS_H_OFF + wave * LDS_H_PER_WAVE;

    // lane's fixed part of the W1P address; per-kk stride = 2*W1P_STRIDE floats,
    // fits the ds 16-bit immediate offset -> clause of ds_load_b64
    const float* w1base = lds + LDS_W1_OFF + hi * W1P_STRIDE + 2 * laneM;

    for (int np = 0; np < 4; ++np) {       // N-tile pairs: two independent acc chains
        const int n0 = 2 * np;
        const float* p0 = w1base + 2 * (n0 * 16);
        const float* p1 = p0 + 32;         // next N-tile: +16 columns

        v2f bf0[16], bf1[16];
        #pragma unroll
        for (int kk = 0; kk < 16; ++kk) {
            bf0[kk] = lds_v2f(p0 + kk * 2 * W1P_STRIDE);
            bf1[kk] = lds_v2f(p1 + kk * 2 * W1P_STRIDE);
        }

        v8f acc0 = {}, acc1 = {};
        #pragma unroll
        for (int kk = 0; kk < 16; ++kk) {
            acc0 = __builtin_amdgcn_wmma_f32_16x16x4_f32(
                false, a1[kk], false, bf0[kk], (short)0, acc0, false, false);
            acc1 = __builtin_amdgcn_wmma_f32_16x16x4_f32(
                false, a1[kk], false, bf1[kk], (short)0, acc1, false, false);
        }

        // fused bias + relu, stash h tiles (C/D layout: M = r + 8*hi, N = n*16+laneM)
        const float bias0 = lds[LDS_B1_OFF + n0 * 16 + laneM];
        const float bias1 = lds[LDS_B1_OFF + n0 * 16 + 16 + laneM];
        #pragma unroll
        for (int r = 0; r < 8; ++r) {
            float v0 = acc0[r] + bias0;  v0 = v0 > 0.0f ? v0 : 0.0f;
            float v1 = acc1[r] + bias1;  v1 = v1 > 0.0f ? v1 : 0.0f;
            hw[(r + 8 * hi) * H_STRIDE + n0 * 16 + laneM]      = v0;
            hw[(r + 8 * hi) * H_STRIDE + n0 * 16 + 16 + laneM] = v1;
        }
    }

    // ---- GEMM2: w[16x32] = h[16x128] @ W2[128x32] + b2 ----
    const float* w2base = lds + LDS_W2_OFF + hi * W2P_STRIDE + 2 * laneM;
    const float* habase = hw + laneM * H_STRIDE + 2 * hi;

    v8f acc2[2] = {{}, {}};
    for (int kb = 0; kb < 32; kb += 8) {   // K blocks of 8: preload, then dense WMMA
        v2f a2f[8], bt0[8], bt1[8];
        #pragma unroll
        for (int j = 0; j < 8; ++j) {
            const int kk = kb + j;
            a2f[j] = lds_v2f(habase + 4 * kk);
            bt0[j] = lds_v2f(w2base + kk * 2 * W2P_STRIDE);        // t=0 (cols 0..15)
            bt1[j] = lds_v2f(w2base + kk * 2 * W2P_STRIDE + 32);   // t=1 (cols 16..31)
        }
        #pragma unroll
        for (int j = 0; j < 8; ++j) {
            acc2[0] = __builtin_amdgcn_wmma_f32_16x16x4_f32(
                false, a2f[j], false, bt0[j], (short)0, acc2[0], false, false);
            acc2[1] = __builtin_amdgcn_wmma_f32_16x16x4_f32(
                false, a2f[j], false, bt1[j], (short)0, acc2[1], false, false);
        }
    }

    // w tile -> LDS (overlay dead h region; per-wave LDS ops are in-order)
    float* ww = hw;
    #pragma unroll
    for (int t = 0; t < 2; ++t) {
        const float bias = lds[LDS_B2_OFF + t * 16 + laneM];
        #pragma unroll
        for (int r = 0; r < 8; ++r)
            ww[(r + 8 * hi) * WW_STRIDE + t * 16 + laneM] = acc2[t][r] + bias;
    }

    // ---- Epilogue: complex contraction out[e,o] = sum_k x[e,k] * w[e,o,k] ----
    // lane -> (edge = laneM, o-pair = hi): o in {2*hi, 2*hi+1}
    const int eOut = eBase + laneM;
    const int og   = hi;

    const float* wrow = ww + laneM * WW_STRIDE + og * 16;   // 16B aligned
    const v4f wq0 = *(const v4f*)__builtin_assume_aligned(wrow + 0, 16);
    const v4f wq1 = *(const v4f*)__builtin_assume_aligned(wrow + 4, 16);
    const v4f wq2 = *(const v4f*)__builtin_assume_aligned(wrow + 8, 16);
    const v4f wq3 = *(const v4f*)__builtin_assume_aligned(wrow + 12, 16);
    float wv[16];
    #pragma unroll
    for (int j = 0; j < 4; ++j) {
        wv[j] = wq0[j]; wv[4 + j] = wq1[j]; wv[8 + j] = wq2[j]; wv[12 + j] = wq3[j];
    }

    int eC = eOut > E - 1 ? E - 1 : eOut;
    const float* xr = x + (long long)eC * 8;                // 32B aligned
    const v4f xq0 = *(const v4f*)(xr);
    const v4f xq1 = *(const v4f*)(xr + 4);
    float xv[8];
    #pragma unroll
    for (int j = 0; j < 4; ++j) { xv[j] = xq0[j]; xv[4 + j] = xq1[j]; }

    float re0 = 0.f, im0 = 0.f, re1 = 0.f, im1 = 0.f;
    #pragma unroll
    for (int k = 0; k < 4; ++k) {
        const float a = xv[2 * k], bb = xv[2 * k + 1];
        re0 += a * wv[2 * k]     - bb * wv[2 * k + 1];
        im0 += a * wv[2 * k + 1] + bb * wv[2 * k];
        re1 += a * wv[8 + 2 * k]     - bb * wv[8 + 2 * k + 1];
        im1 += a * wv[8 + 2 * k + 1] + bb * wv[8 + 2 * k];
    }

    if (eOut < E) {
        v4f o4; o4[0] = re0; o4[1] = im0; o4[2] = re1; o4[3] = im1;
        *(v4f*)(out + (long long)eOut * 8 + og * 4) = o4;   // 16B aligned
    }
}

extern "C" void kernel_launch(void* const* d_in, const int* in_sizes, int n_in,
                              void* d_out, int out_size, void* d_ws, size_t ws_size,
                              hipStream_t stream) {
    const float* x  = (const float*)d_in[0];
    const float* s  = (const float*)d_in[1];
    const float* W1 = (const float*)d_in[2];
    const float* b1 = (const float*)d_in[3];
    const float* W2 = (const float*)d_in[4];
    const float* b2 = (const float*)d_in[5];
    float* out = (float*)d_out;

    const int E = in_sizes[1] / 64;   // s is (E, 64)
    const int blocks = (E + EDGES_PER_BLOCK - 1) / EDGES_PER_BLOCK;
    const size_t shmem = (size_t)LDS_FLOATS * sizeof(float);

    so2_linear_escnn_kernel<<<blocks, 256, shmem, stream>>>(x, s, W1, b1, W2, b2, out, E);
}